// GNN_12043088298451
// MI455X (gfx1250) — compile-verified
//
#include <hip/hip_runtime.h>
#include <hip/hip_bf16.h>

typedef __attribute__((ext_vector_type(2))) float v2f;
typedef __attribute__((ext_vector_type(8))) float v8f;

#define DIM 32

__device__ __forceinline__ void atomic_add_f32_fast(float* p, float v) {
  // lowers to global_atomic_add_f32 (no CAS loop) on CDNA
  unsafeAtomicAdd(p, v);
}

// ---------------------------------------------------------------------------
// 0) gather x = entity_emb[node_ids], init deg = 1.0 (self loop)
__global__ void k_gather_init(const float* __restrict__ emb, const int* __restrict__ ids,
                              float* __restrict__ x, float* __restrict__ deg, int N) {
  int tid = blockIdx.x * blockDim.x + threadIdx.x;
  if (tid < N) deg[tid] = 1.0f;
  if (tid < N * DIM) {
    int n = tid >> 5, d = tid & 31;
    x[tid] = emb[(size_t)ids[n] * DIM + d];
  }
}

// 1) deg[dst[e]] += 1
__global__ void k_deg(const int* __restrict__ dst, float* __restrict__ deg, int E) {
  int e = blockIdx.x * blockDim.x + threadIdx.x;
  if (e < E) atomic_add_f32_fast(&deg[dst[e]], 1.0f);
}

// 2) dis = rsqrt(deg), in place
__global__ void k_rsqrt(float* __restrict__ deg, int N) {
  int n = blockIdx.x * blockDim.x + threadIdx.x;
  if (n < N) deg[n] = rsqrtf(deg[n]);
}

// ---------------------------------------------------------------------------
// WMMA GEMM: O[nrows,32] = A[nrows,32] @ W[32,32] (+bias, relu optional)
// One wave handles 16 rows -> two 16x16 tiles, K=32 in 8 steps of 16x16x4 f32.
__global__ void k_gemm32(const float* __restrict__ A, const float* __restrict__ W,
                         float* __restrict__ O, int nrows,
                         const float* __restrict__ bias, int do_relu) {
  const int lane = threadIdx.x & 31;
  const int wave = blockIdx.x * (blockDim.x >> 5) + (threadIdx.x >> 5);
  const int row0 = wave * 16;
  if (row0 >= nrows) return;           // wave-uniform exit: EXEC stays all-1s for WMMA

  const int half = lane >> 4;          // 0: K pair {0,1}; 1: K pair {2,3} (per 4-wide K step)
  const int l15  = lane & 15;
  const float* __restrict__ arow = A + (size_t)(row0 + l15) * DIM;

  v8f c0 = {0.f,0.f,0.f,0.f,0.f,0.f,0.f,0.f};
  v8f c1 = {0.f,0.f,0.f,0.f,0.f,0.f,0.f,0.f};

#pragma unroll
  for (int k = 0; k < 8; ++k) {
    const int kk = k * 4 + half * 2;
    // A 16x4 fragment: lane l15 holds row row0+l15, K = kk, kk+1 (8B aligned load)
    v2f a = *(const v2f*)(arow + kk);
    // B 4x16 fragments (tile 0: cols 0-15, tile 1: cols 16-31): lane = column
    v2f b0, b1;
    b0.x = W[(size_t)kk * DIM + l15];
    b0.y = W[(size_t)(kk + 1) * DIM + l15];
    b1.x = W[(size_t)kk * DIM + 16 + l15];
    b1.y = W[(size_t)(kk + 1) * DIM + 16 + l15];
    c0 = __builtin_amdgcn_wmma_f32_16x16x4_f32(false, a, false, b0, (short)0, c0, false, false);
    c1 = __builtin_amdgcn_wmma_f32_16x16x4_f32(false, a, false, b1, (short)0, c1, false, false);
  }

  // C/D layout: lane(0-15) = column, v[r] = row (lanes 16-31: rows 8-15)
  float bv0 = 0.f, bv1 = 0.f;
  if (bias) { bv0 = bias[l15]; bv1 = bias[16 + l15]; }
#pragma unroll
  for (int r = 0; r < 8; ++r) {
    const int m = half * 8 + r;
    float v0 = c0[r] + bv0;
    float v1 = c1[r] + bv1;
    if (do_relu) { v0 = fmaxf(v0, 0.f); v1 = fmaxf(v1, 0.f); }
    float* orow = O + (size_t)(row0 + m) * DIM;
    orow[l15]      = v0;
    orow[16 + l15] = v1;
  }
}

// ---------------------------------------------------------------------------
// 3) self-loop contribution: agg[n,:] = dis[n]^2 * h[n,:]
__global__ void k_self(const float* __restrict__ h, const float* __restrict__ dis,
                       float* __restrict__ agg, int total) {
  int gid = blockIdx.x * blockDim.x + threadIdx.x;
  if (gid >= total) return;
  float dn = dis[gid >> 5];
  agg[gid] = dn * dn * h[gid];
}

// 4) edge scatter: one wave per edge, lane = feature (D == wave32)
__global__ void k_edge_scatter(const int* __restrict__ src, const int* __restrict__ dst,
                               const float* __restrict__ h, const float* __restrict__ dis,
                               float* __restrict__ agg, int E) {
  int gid = blockIdx.x * blockDim.x + threadIdx.x;
  int e = gid >> 5, d = gid & 31;
  if (e >= E) return;
  int s = src[e], t = dst[e];
  float norm = dis[s] * dis[t];
  atomic_add_f32_fast(&agg[(size_t)t * DIM + d], h[(size_t)s * DIM + d] * norm);
}

// 5) x = relu(agg + b)
__global__ void k_bias_relu(const float* __restrict__ agg, const float* __restrict__ b,
                            float* __restrict__ x, int total) {
  int gid = blockIdx.x * blockDim.x + threadIdx.x;
  if (gid < total) x[gid] = fmaxf(agg[gid] + b[gid & 31], 0.f);
}

// 6) triple[t,:] = x[head[t],:] + rel_emb[rel[t],:] + x[tail[t],:]
__global__ void k_triple(const float* __restrict__ x, const float* __restrict__ rel_emb,
                         const int* __restrict__ head, const int* __restrict__ tail,
                         const int* __restrict__ rel, float* __restrict__ triple, int T) {
  int gid = blockIdx.x * blockDim.x + threadIdx.x;
  int t = gid >> 5, d = gid & 31;
  if (t >= T) return;
  triple[gid] = x[(size_t)head[t] * DIM + d]
              + rel_emb[(size_t)rel[t] * DIM + d]
              + x[(size_t)tail[t] * DIM + d];
}

// 7) out[t,0:3] = th[t,:] @ Wm2[32,3] + bm2
__global__ void k_head(const float* __restrict__ th, const float* __restrict__ Wm2,
                       const float* __restrict__ bm2, float* __restrict__ out, int T) {
  int t = blockIdx.x * blockDim.x + threadIdx.x;
  if (t >= T) return;
  const float* r = th + (size_t)t * DIM;
  float o0 = bm2[0], o1 = bm2[1], o2 = bm2[2];
#pragma unroll
  for (int d = 0; d < DIM; ++d) {
    float v = r[d];
    o0 += v * Wm2[d * 3 + 0];
    o1 += v * Wm2[d * 3 + 1];
    o2 += v * Wm2[d * 3 + 2];
  }
  out[(size_t)t * 3 + 0] = o0;
  out[(size_t)t * 3 + 1] = o1;
  out[(size_t)t * 3 + 2] = o2;
}

// ---------------------------------------------------------------------------
extern "C" void kernel_launch(void* const* d_in, const int* in_sizes, int n_in,
                              void* d_out, int out_size, void* d_ws, size_t ws_size,
                              hipStream_t stream) {
  const float* entity_emb   = (const float*)d_in[0];
  const float* relation_emb = (const float*)d_in[1];
  const float* W1  = (const float*)d_in[2];
  const float* b1  = (const float*)d_in[3];
  const float* W2  = (const float*)d_in[4];
  const float* b2  = (const float*)d_in[5];
  const float* Wm1 = (const float*)d_in[6];
  const float* bm1 = (const float*)d_in[7];
  const float* Wm2 = (const float*)d_in[8];
  const float* bm2 = (const float*)d_in[9];
  const int* node_ids = (const int*)d_in[10];
  const int* src      = (const int*)d_in[11];
  const int* dst      = (const int*)d_in[12];
  const int* head_idx = (const int*)d_in[13];
  const int* tail_idx = (const int*)d_in[14];
  const int* rel_idx  = (const int*)d_in[15];

  const int N = in_sizes[10];
  const int E = in_sizes[11];
  const int T = in_sizes[13];

  // workspace layout (floats): dis[N] | x[N*32] | big = max(h[N*32]+agg[N*32], triple[T*32])
  float* ws  = (float*)d_ws;
  float* dis = ws;
  float* x   = ws + N;
  float* big = x + (size_t)N * DIM;
  float* h   = big;
  float* agg = big + (size_t)N * DIM;
  float* triple = big;          // reused after GCN layers complete

  const dim3 blk(256);
  auto nblk = [](long long t) { return dim3((unsigned)((t + 255) / 256)); };

  k_gather_init<<<nblk((long long)N * DIM), blk, 0, stream>>>(entity_emb, node_ids, x, dis, N);
  k_deg<<<nblk(E), blk, 0, stream>>>(dst, dis, E);
  k_rsqrt<<<nblk(N), blk, 0, stream>>>(dis, N);

  const long long nwaves = ((long long)N + 15) / 16;
  for (int layer = 0; layer < 2; ++layer) {
    const float* W = (layer == 0) ? W1 : W2;
    const float* b = (layer == 0) ? b1 : b2;
    k_gemm32<<<nblk(nwaves * 32), blk, 0, stream>>>(x, W, h, N, nullptr, 0);
    k_self<<<nblk((long long)N * DIM), blk, 0, stream>>>(h, dis, agg, N * DIM);
    k_edge_scatter<<<nblk((long long)E * 32), blk, 0, stream>>>(src, dst, h, dis, agg, E);
    k_bias_relu<<<nblk((long long)N * DIM), blk, 0, stream>>>(agg, b, x, N * DIM);
  }

  k_triple<<<nblk((long long)T * DIM), blk, 0, stream>>>(x, relation_emb, head_idx, tail_idx,
                                                         rel_idx, triple, T);
  const long long twaves = ((long long)T + 15) / 16;
  k_gemm32<<<nblk(twaves * 32), blk, 0, stream>>>(triple, Wm1, triple, T, bm1, 1);
  k_head<<<nblk(T), blk, 0, stream>>>(triple, Wm2, bm2, (float*)d_out, T);
}